// GCN_28054726377560
// MI455X (gfx1250) — compile-verified
//
#include <hip/hip_runtime.h>

typedef __attribute__((ext_vector_type(2))) float v2f;
typedef __attribute__((ext_vector_type(8))) float v8f;

// ---------------------------------------------------------------------------
// Degree / normalization
// ---------------------------------------------------------------------------
__global__ void deg_init_kernel(float* __restrict__ deg, int n) {
  int i = blockIdx.x * blockDim.x + threadIdx.x;
  if (i < n) deg[i] = 1.0f;  // self-loop contributes 1 to every node's degree
}

__global__ void deg_accum_kernel(const long long* __restrict__ dst,
                                 float* __restrict__ deg, int E) {
  int e = blockIdx.x * blockDim.x + threadIdx.x;
  if (e < E) atomicAdd(&deg[dst[e]], 1.0f);
}

__global__ void dinv_kernel(const float* __restrict__ deg,
                            float* __restrict__ dinv, int n) {
  int i = blockIdx.x * blockDim.x + threadIdx.x;
  if (i < n) {
    float d = deg[i];
    dinv[i] = (d > 0.0f) ? rsqrtf(d) : 0.0f;
  }
}

// ---------------------------------------------------------------------------
// hs = dinv[row] * (X[nrows x 128] @ W[128 x NOUT]); also acc = hs (self-loop
// initialized accumulator). fp32 WMMA 16x16x4, one wave per 16-row tile,
// NOUT/16 column tiles accumulated in registers.
//
// W is staged in LDS pre-swizzled into K-pair-packed float2s:
//   sW[kp * NOUT + n] = ( W[2*kp][n], W[2*kp+1][n] )
// so each lane's B-fragment is ONE aligned ds_load_b64 straight into an
// adjacent VGPR pair (no v_mov shuffles feeding the WMMA).
// ---------------------------------------------------------------------------
template <int NOUT>
__global__ __launch_bounds__(256) void gemm_scaled_kernel(
    const float* __restrict__ X, const float* __restrict__ W,
    const float* __restrict__ dinv, float* __restrict__ hs,
    float* __restrict__ acc, int nrows) {
  __shared__ v2f sW[64 * NOUT];  // 128*NOUT floats, K-pair packed
  for (int i = threadIdx.x; i < 64 * NOUT; i += 256) {
    const int kp = i / NOUT;  // K pair index: rows 2*kp, 2*kp+1
    const int nn = i % NOUT;
    v2f w;
    w.x = W[(2 * kp) * NOUT + nn];      // coalesced across threads (nn varies)
    w.y = W[(2 * kp + 1) * NOUT + nn];
    sW[i] = w;                          // conflict-free ds_store_b64
  }
  __syncthreads();

  const int wave = threadIdx.x >> 5;
  const int lane = threadIdx.x & 31;
  const int tile = blockIdx.x * 8 + wave;
  if (tile * 16 >= nrows) return;  // wave-uniform exit (after barrier)

  const int row0 = tile * 16;
  const int ln = lane & 15;
  const int kk = (lane < 16) ? 0 : 2;  // 16x4 f32 A-frag: hi half-wave holds K+2

  constexpr int NT = NOUT / 16;
  v8f accf[NT];
#pragma unroll
  for (int t = 0; t < NT; ++t) accf[t] = (v8f)(0.0f);

  const float* xrow = X + (size_t)(row0 + ln) * 128;

#pragma unroll 4
  for (int k = 0; k < 128; k += 4) {
    const v2f a = *(const v2f*)(xrow + k + kk);  // one global_load_b64
    const int kp = (k + kk) >> 1;
#pragma unroll
    for (int t = 0; t < NT; ++t) {
      const v2f b = sW[kp * NOUT + t * 16 + ln];  // one ds_load_b64
      accf[t] = __builtin_amdgcn_wmma_f32_16x16x4_f32(
          false, a, false, b, (short)0, accf[t], false, false);
    }
  }

  // C/D layout: VGPR i -> row (row0 + i + (lane<16?0:8)), col = t*16 + ln
  const int rbase = row0 + ((lane < 16) ? 0 : 8);
#pragma unroll
  for (int i = 0; i < 8; ++i) {
    const int r = rbase + i;
    const float s = dinv[r];
#pragma unroll
    for (int t = 0; t < NT; ++t) {
      float v = accf[t][i] * s;
      hs[(size_t)r * NOUT + t * 16 + ln] = v;
      acc[(size_t)r * NOUT + t * 16 + ln] = v;
    }
  }
}

// ---------------------------------------------------------------------------
// Edge scatter: acc[dst, :] += hs[src, :]. One wave per edge, lanes split the
// row -> coalesced b128/b64 gather + fp32 global atomics (L2-resident table).
// ---------------------------------------------------------------------------
template <int C>
__global__ __launch_bounds__(256) void edge_scatter_kernel(
    const long long* __restrict__ src, const long long* __restrict__ dst,
    const float* __restrict__ hs, float* __restrict__ acc, int E) {
  const int gw = (int)((blockIdx.x * blockDim.x + threadIdx.x) >> 5);
  const int lane = threadIdx.x & 31;
  if (gw >= E) return;  // wave-uniform
  const long long s = src[gw];
  const long long d = dst[gw];
  constexpr int F = C / 32;
  const float* hp = hs + (size_t)s * C + lane * F;
  float* ap = acc + (size_t)d * C + lane * F;
  float v[F];
#pragma unroll
  for (int j = 0; j < F; ++j) v[j] = hp[j];  // vectorizes to b128/b64 load
#pragma unroll
  for (int j = 0; j < F; ++j) atomicAdd(&ap[j], v[j]);
}

// ---------------------------------------------------------------------------
// out = relu(dinv[row] * acc + bias[col])
// ---------------------------------------------------------------------------
template <int C>
__global__ void bias_relu_kernel(const float* __restrict__ acc,
                                 const float* __restrict__ dinv,
                                 const float* __restrict__ bias,
                                 float* __restrict__ out, int n) {
  const int idx = blockIdx.x * blockDim.x + threadIdx.x;
  if (idx >= n * C) return;
  const int r = idx / C;
  const int c = idx % C;
  float v = dinv[r] * acc[idx] + bias[c];
  out[idx] = (v > 0.0f) ? v : 0.0f;
}

// ---------------------------------------------------------------------------
extern "C" void kernel_launch(void* const* d_in, const int* in_sizes, int n_in,
                              void* d_out, int out_size, void* d_ws,
                              size_t ws_size, hipStream_t stream) {
  (void)n_in; (void)out_size; (void)ws_size;

  const float* x = (const float*)d_in[0];
  const long long* ei = (const long long*)d_in[1];  // int64 [2, E]
  const float* W1 = (const float*)d_in[2];
  const float* b1 = (const float*)d_in[3];
  const float* W2 = (const float*)d_in[4];
  const float* b2 = (const float*)d_in[5];

  const int n = in_sizes[0] / 128;  // 50000
  const int E = in_sizes[1] / 2;    // 800000
  const long long* srcp = ei;
  const long long* dstp = ei + E;

  float* ws = (float*)d_ws;
  float* deg = ws;                              // n
  float* dinv = deg + n;                        // n
  float* h1s = dinv + n;                        // n*128
  float* acc1 = h1s + (size_t)n * 128;          // n*128
  // layer-2 buffers alias the retired h1s region (disjoint from acc1):
  float* h2s = h1s;                             // n*64
  float* acc2 = h1s + (size_t)n * 64;           // n*64

  const int tiles = (n + 15) / 16;
  const int gemm_blocks = (tiles + 7) / 8;
  const int edge_blocks = (E + 7) / 8;  // 8 waves/block, wave per edge

  // Degree + normalization
  deg_init_kernel<<<(n + 255) / 256, 256, 0, stream>>>(deg, n);
  deg_accum_kernel<<<(E + 255) / 256, 256, 0, stream>>>(dstp, deg, E);
  dinv_kernel<<<(n + 255) / 256, 256, 0, stream>>>(deg, dinv, n);

  // Layer 1: h1s = dinv*(x@W1); acc1 = h1s; scatter; relu(dinv*acc1+b1) -> acc1
  gemm_scaled_kernel<128><<<gemm_blocks, 256, 0, stream>>>(x, W1, dinv, h1s,
                                                           acc1, n);
  edge_scatter_kernel<128><<<edge_blocks, 256, 0, stream>>>(srcp, dstp, h1s,
                                                            acc1, E);
  bias_relu_kernel<128><<<(n * 128 + 255) / 256, 256, 0, stream>>>(
      acc1, dinv, b1, acc1, n);

  // Layer 2: h2s = dinv*(acc1@W2); acc2 = h2s; scatter; relu -> d_out
  gemm_scaled_kernel<64><<<gemm_blocks, 256, 0, stream>>>(acc1, W2, dinv, h2s,
                                                          acc2, n);
  edge_scatter_kernel<64><<<edge_blocks, 256, 0, stream>>>(srcp, dstp, h2s,
                                                           acc2, E);
  bias_relu_kernel<64><<<(n * 64 + 255) / 256, 256, 0, stream>>>(
      acc2, dinv, b2, (float*)d_out, n);
}